// BatchDynamicSoftLabelAssigner_77996606095747
// MI455X (gfx1250) — compile-verified
//
#include <hip/hip_runtime.h>
#include <hip/hip_bf16.h>

#define BATCH 4
#define NPRI  8400
#define NGT   32
#define NCLS  80
#define INF_COST 100000000.0f

typedef __attribute__((ext_vector_type(2))) float v2f;
typedef __attribute__((ext_vector_type(8))) float v8f;

__device__ __forceinline__ float sigmoidf(float l) {
    return 1.0f / (1.0f + __expf(-l));
}
__device__ __forceinline__ float softplusf(float l) {
    // stable: max(l,0) + log1p(exp(-|l|))
    return fmaxf(l, 0.0f) + log1pf(expf(-fabsf(l)));
}
__device__ __forceinline__ float gfun(float l) {
    float s = sigmoidf(l);
    return softplusf(l) * s * s;
}

// ---------------------------------------------------------------------------
// Kernel A: one wave (32 threads) handles a tile of 16 priors for one batch.
//   - base[i] = sum_c softplus(l)*sigmoid(l)^2 via V_WMMA_F32_16X16X4_F32
//     (A = 16 priors x 4 classes of g(l), B = ones -> D column holds row sums)
//   - validity, per-pair cost & IoU written transposed (b, j, p)
// ---------------------------------------------------------------------------
__global__ void __launch_bounds__(32)
assign_cost_kernel(const float* __restrict__ pred_bboxes,
                   const float* __restrict__ pred_scores,
                   const float* __restrict__ priors,
                   const int*   __restrict__ gt_labels,
                   const float* __restrict__ gt_bboxes,
                   const float* __restrict__ flag,
                   float* __restrict__ costT,
                   float* __restrict__ iouT,
                   unsigned* __restrict__ matchmask)
{
    const int b    = blockIdx.y;
    const int tile = blockIdx.x * 16;
    const int lane = threadIdx.x;      // 0..31, full wave active (WMMA needs EXEC all-1s)
    const int m    = lane & 15;        // prior row within tile (matches WMMA A-layout M=lane%16)
    const int half = lane >> 4;        // 0: K slots 0,1   1: K slots 2,3
    const int p    = tile + m;

    const float* sc = pred_scores + ((size_t)(b * NPRI) + p) * NCLS;

    // ---- base[i] over 80 classes with WMMA f32 16x16x4, B = ones ----
    v8f acc = {};
    v2f bones; bones.x = 1.0f; bones.y = 1.0f;
    #pragma unroll
    for (int c0 = 0; c0 < NCLS; c0 += 4) {
        const v2f lpair = *(const v2f*)(sc + c0 + 2 * half);   // b64 load
        v2f a; a.x = gfun(lpair.x); a.y = gfun(lpair.y);
        acc = __builtin_amdgcn_wmma_f32_16x16x4_f32(
            false, a, false, bones, (short)0, acc, false, false);
    }
    // D[m,n] identical across n (B==ones). Column 0: lane0 rows 0..7, lane16 rows 8..15.
    __shared__ float lbase[16];
    if (lane == 0) {
        #pragma unroll
        for (int r = 0; r < 8; ++r) lbase[r] = acc[r];
    }
    if (lane == 16) {
        #pragma unroll
        for (int r = 0; r < 8; ++r) lbase[8 + r] = acc[r];
    }
    __syncthreads();
    const float base = lbase[m];

    // ---- prior / pred box data ----
    const float px = priors[p * 4 + 0];
    const float py = priors[p * 4 + 1];
    const float st = priors[p * 4 + 2];

    const float* pb = pred_bboxes + ((size_t)(b * NPRI) + p) * 4;
    const float bx1 = pb[0], by1 = pb[1], bx2 = pb[2], by2 = pb[3];
    const float a1  = (bx2 - bx1) * (by2 - by1);

    // ---- validity: prior center strictly inside any real gt box ----
    int inAny = 0;
    for (int t = 0; t < 16; ++t) {
        const int j = half * 16 + t;
        const float* gb = gt_bboxes + (size_t)(b * NGT + j) * 4;
        const float fl  = flag[b * NGT + j];
        float mn = fminf(fminf(px - gb[0], py - gb[1]),
                         fminf(gb[2] - px, gb[3] - py));
        if (mn > 0.0f && fl > 0.0f) inAny = 1;
    }
    inAny |= __shfl_xor(inAny, 16, 32);
    const bool valid = (inAny != 0);

    if (half == 0) matchmask[b * NPRI + p] = 0u;   // init for kernel B's atomicOr

    // ---- per (prior, gt) cost & IoU; this lane covers 16 gts ----
    for (int t = 0; t < 16; ++t) {
        const int j = half * 16 + t;
        const float* gb = gt_bboxes + (size_t)(b * NGT + j) * 4;
        const float gx1 = gb[0], gy1 = gb[1], gx2 = gb[2], gy2 = gb[3];

        // IoU (union clamped at 1e-6, as reference)
        const float iw = fmaxf(fminf(bx2, gx2) - fmaxf(bx1, gx1), 0.0f);
        const float ih = fmaxf(fminf(by2, gy2) - fmaxf(by1, gy1), 0.0f);
        const float inter = iw * ih;
        const float a2 = (gx2 - gx1) * (gy2 - gy1);
        const float iou = inter / fmaxf(a1 + a2 - inter, 1e-6f);

        // soft center prior: 10^(dist/stride - 3); dist zeroed when prior invalid
        const float dx = px - 0.5f * (gx1 + gx2);
        const float dy = py - 0.5f * (gy1 + gy2);
        float dist = sqrtf(dx * dx + dy * dy) / st;
        if (!valid) dist = 0.0f;
        const float scp = exp2f(3.3219280948873623f * (dist - 3.0f));

        const float iou_cost = -logf(iou + 1e-7f) * 3.0f;

        // class cost via factorization: base - g(l_lab) + (sp - l*iou)*(iou - sig)^2
        const int   lab = gt_labels[b * NGT + j];
        const float l   = sc[lab];
        const float sig = sigmoidf(l);
        const float sp  = softplusf(l);
        const float dif = iou - sig;
        const float cls = base - sp * sig * sig + (sp - l * iou) * dif * dif;

        float cost = cls + iou_cost + scp;
        if (!valid) cost = INF_COST;

        const size_t o = (size_t)(b * NGT + j) * NPRI + p;
        costT[o] = cost;
        iouT[o]  = iou;
    }
}

// ---------------------------------------------------------------------------
// Kernel B: one block per (gt j, batch b). dynamic_k from top-13 IoU sum,
// then k stable-order argmin passes over cost -> set match bit j per prior.
// ---------------------------------------------------------------------------
__global__ void __launch_bounds__(256)
topk_kernel(const float* __restrict__ costT,
            const float* __restrict__ iouT,
            const float* __restrict__ flag,
            unsigned* __restrict__ matchmask)
{
    const int j = blockIdx.x;
    const int b = blockIdx.y;
    if (flag[b * NGT + j] <= 0.0f) return;   // padded gt: no matches

    const float* iouc  = iouT  + (size_t)(b * NGT + j) * NPRI;
    const float* costc = costT + (size_t)(b * NGT + j) * NPRI;
    const int tid = threadIdx.x;

    __shared__ float sv[256];
    __shared__ int   si[256];

    // ---- sum of top-13 IoUs (strict (value desc, index asc) order) ----
    float lastV = 3.0e38f; int lastI = -1;
    float sum = 0.0f;
    for (int t = 0; t < 13; ++t) {
        float bv = -2.0f; int bi = NPRI;
        for (int p = tid; p < NPRI; p += 256) {
            const float v = iouc[p];
            const bool cand = (v < lastV) || (v == lastV && p > lastI);
            if (cand && (v > bv || (v == bv && p < bi))) { bv = v; bi = p; }
        }
        sv[tid] = bv; si[tid] = bi; __syncthreads();
        for (int s = 128; s > 0; s >>= 1) {
            if (tid < s) {
                const float ov = sv[tid + s]; const int oi = si[tid + s];
                if (ov > sv[tid] || (ov == sv[tid] && oi < si[tid])) { sv[tid] = ov; si[tid] = oi; }
            }
            __syncthreads();
        }
        lastV = sv[0]; lastI = si[0];
        __syncthreads();
        sum += lastV;
    }
    int k = (int)sum;                // trunc toward zero, sum >= 0
    if (k < 1)  k = 1;
    if (k > 13) k = 13;

    // ---- k passes: stable argmin over cost ((value asc, index asc)) ----
    lastV = -3.0e38f; lastI = -1;
    for (int t = 0; t < k; ++t) {
        float bv = 3.0e38f; int bi = NPRI;
        for (int p = tid; p < NPRI; p += 256) {
            const float v = costc[p];
            const bool cand = (v > lastV) || (v == lastV && p > lastI);
            if (cand && (v < bv || (v == bv && p < bi))) { bv = v; bi = p; }
        }
        sv[tid] = bv; si[tid] = bi; __syncthreads();
        for (int s = 128; s > 0; s >>= 1) {
            if (tid < s) {
                const float ov = sv[tid + s]; const int oi = si[tid + s];
                if (ov < sv[tid] || (ov == sv[tid] && oi < si[tid])) { sv[tid] = ov; si[tid] = oi; }
            }
            __syncthreads();
        }
        lastV = sv[0]; lastI = si[0];
        __syncthreads();
        if (tid == 0 && lastI < NPRI)
            atomicOr(&matchmask[b * NPRI + lastI], 1u << j);
    }
}

// ---------------------------------------------------------------------------
// Kernel C: per prior — conflict resolution + gather outputs.
// Output (floats, concatenated): labels[B*N], weights[B*N], boxes[B*N*4], metrics[B*N]
// ---------------------------------------------------------------------------
__global__ void __launch_bounds__(256)
finalize_kernel(const float* __restrict__ costT,
                const float* __restrict__ iouT,
                const int*   __restrict__ gt_labels,
                const float* __restrict__ gt_bboxes,
                const unsigned* __restrict__ matchmask,
                float* __restrict__ out)
{
    const int idx = blockIdx.x * blockDim.x + threadIdx.x;
    if (idx >= BATCH * NPRI) return;
    const int b = idx / NPRI;
    const int p = idx - b * NPRI;

    const unsigned msk = matchmask[idx];
    const int cnt = __popc(msk);

    int jm = 0; bool fg = false; float miou = 0.0f;
    if (cnt == 1) {
        jm = __ffs(msk) - 1;           // argmax of one-hot row
        fg = true;
        miou = iouT[(size_t)(b * NGT + jm) * NPRI + p];
    } else if (cnt > 1) {
        // conflict: argmin_j cost (first occurrence), over ALL 32 gts
        float bestc = 3.4e38f;
        for (int j = 0; j < NGT; ++j) {
            const float c = costT[(size_t)(b * NGT + j) * NPRI + p];
            if (c < bestc) { bestc = c; jm = j; }
        }
        fg = true;
        miou = iouT[(size_t)(b * NGT + jm) * NPRI + p];
    }

    float* labels  = out;
    float* weights = out + (size_t)BATCH * NPRI;
    float* boxes   = out + (size_t)2 * BATCH * NPRI;
    float* metrics = out + (size_t)6 * BATCH * NPRI;

    labels[idx]  = fg ? (float)gt_labels[b * NGT + jm] : (float)NCLS;
    weights[idx] = 1.0f;
    if (fg) {
        const float* gb = gt_bboxes + (size_t)(b * NGT + jm) * 4;
        #pragma unroll
        for (int t = 0; t < 4; ++t) boxes[(size_t)idx * 4 + t] = gb[t];
    } else {
        #pragma unroll
        for (int t = 0; t < 4; ++t) boxes[(size_t)idx * 4 + t] = 0.0f;
    }
    metrics[idx] = fg ? miou : 0.0f;
}

extern "C" void kernel_launch(void* const* d_in, const int* in_sizes, int n_in,
                              void* d_out, int out_size, void* d_ws, size_t ws_size,
                              hipStream_t stream) {
    const float* pred_bboxes = (const float*)d_in[0];
    const float* pred_scores = (const float*)d_in[1];
    const float* priors      = (const float*)d_in[2];
    const int*   gt_labels   = (const int*)  d_in[3];
    const float* gt_bboxes   = (const float*)d_in[4];
    const float* flag        = (const float*)d_in[5];

    // scratch layout: costT[B*G*N] | iouT[B*G*N] | matchmask[B*N] (~8.7 MB)
    float*    costT     = (float*)d_ws;
    float*    iouT      = costT + (size_t)BATCH * NGT * NPRI;
    unsigned* matchmask = (unsigned*)(iouT + (size_t)BATCH * NGT * NPRI);

    dim3 gA(NPRI / 16, BATCH);
    assign_cost_kernel<<<gA, 32, 0, stream>>>(pred_bboxes, pred_scores, priors,
                                              gt_labels, gt_bboxes, flag,
                                              costT, iouT, matchmask);

    dim3 gB(NGT, BATCH);
    topk_kernel<<<gB, 256, 0, stream>>>(costT, iouT, flag, matchmask);

    const int total = BATCH * NPRI;
    finalize_kernel<<<(total + 255) / 256, 256, 0, stream>>>(costT, iouT, gt_labels,
                                                             gt_bboxes, matchmask,
                                                             (float*)d_out);
}